// CgsGraphConv_83476984365385
// MI455X (gfx1250) — compile-verified
//
#include <hip/hip_runtime.h>
#include <math.h>

#define BB   128
#define NN   36
#define CC   1024
#define KNB  16
#define KER  8
#define DK   128            // OUT / KER
#define BN   (BB*NN)        // 4608
#define MT   16             // nodes per block (one WMMA M-tile)
#define NBLK (BN/MT)        // 288
#define CHUNK  256          // C-chunk held in LDS
#define NCHUNK (CC/CHUNK)   // 4
#define SP   (CHUNK+4)      // padded agg row stride (floats) -> conflict-free ds_load_b64

typedef float v2f __attribute__((ext_vector_type(2)));
typedef float v4f __attribute__((ext_vector_type(4)));
typedef float v8f __attribute__((ext_vector_type(8)));

// dynamic smem layout (in floats):
//   [0,256)                : sm_idx (int)  gathered neighbor indices
//   [256, 256+2048)        : sm_ew  [nd][nb][k]  edge weights (float4-aligned)
//   [2304, 2304+8*16*SP)   : sm_agg [k][nd][SP]  aggregate chunk
#define EW_OFF   256
#define AGG_OFF  (EW_OFF + MT*KNB*KER)          // 2304
#define SMEM_FLOATS (AGG_OFF + KER*MT*SP)       // 35584 floats = 142336 B

extern "C" __global__ __launch_bounds__(256)
void cgs_graphconv_wmma(const float* __restrict__ feats,     // [BN, C]
                        const float* __restrict__ centre,    // [BN, 2]
                        const int*   __restrict__ nidx,      // [BN, KNB]
                        const float* __restrict__ gwt,       // [BN*KNB]
                        const float* __restrict__ mean_rho,  // [8]
                        const float* __restrict__ mean_th,   // [8]
                        const float* __restrict__ prec_rho,  // [8]
                        const float* __restrict__ prec_th,   // [8]
                        const float* __restrict__ convw,     // [KER, DK, C]
                        float*       __restrict__ out)       // [BN, 1024]
{
    extern __shared__ float smem[];
    int*   sm_idx = (int*)smem;
    float* sm_ew  = smem + EW_OFF;
    float* sm_agg = smem + AGG_OFF;

    const int t     = threadIdx.x;        // 0..255
    const int node0 = blockIdx.x * MT;

    // ---------------- Phase 0: edge weights, one edge per thread ----------------
    {
        const int nd   = t >> 4;           // local node 0..15
        const int nb   = t & 15;           // neighbor   0..15
        const int node = node0 + nd;
        const int e    = node * KNB + nb;
        const int j    = nidx[e];
        sm_idx[t] = j;

        const float cx  = centre[node*2+0] - centre[j*2+0];
        const float cy  = centre[node*2+1] - centre[j*2+1];
        const float rho = sqrtf(cx*cx + cy*cy);
        const float th  = atan2f(cx, cy);          // note arg order: (x, y) per reference

        float w[KER];
        float s = 0.f;
        #pragma unroll
        for (int k = 0; k < KER; ++k) {
            const float dr = rho - mean_rho[k];
            const float pr = prec_rho[k];
            const float wr = __expf(-0.5f * dr * dr / (1e-14f + pr * pr));
            const float fa = fabsf(th - mean_th[k]);
            const float sa = fabsf(6.28318530717958648f - fa);
            const float mi = fminf(fa, sa);
            const float pt = prec_th[k];
            const float wt = __expf(-0.5f * mi * mi / (1e-14f + pt * pt));
            float wk = wr * wt;
            wk = (wk != wk) ? 0.f : wk;            // NaN -> 0
            w[k] = wk;
            s += wk;
        }
        const float scale = gwt[e] / s;
        #pragma unroll
        for (int k = 0; k < KER; ++k)
            sm_ew[(nd*KNB + nb)*KER + k] = w[k] * scale;
    }
    __syncthreads();

    // ---------------- WMMA setup: wave w <-> gaussian kernel w ----------------
    const int wave = t >> 5;               // 0..7 == kernel index
    const int lane = t & 31;
    const int colL = lane & 15;            // N column / A row (M)
    const int half = lane >> 4;            // 0 | 1
    const int koff = half * 2;             // K offset for A/B per ISA layout

    v8f acc[8];
    {
        const v8f z = {0.f,0.f,0.f,0.f,0.f,0.f,0.f,0.f};
        #pragma unroll
        for (int dt = 0; dt < 8; ++dt) acc[dt] = z;
    }

    // B element (row=c, col=d) = convw[k][d0+colL][c]; per-lane row pointer
    const float* bbase = convw + (size_t)wave * DK * CC + (size_t)colL * CC + koff;

    for (int cc = 0; cc < NCHUNK; ++cc) {
        const int c0 = cc * CHUNK;

        // ---- aggregate chunk into LDS: thread t owns channel c=t for all nodes ----
        const v4f* ew4 = (const v4f*)sm_ew;
        #pragma unroll 2
        for (int nd = 0; nd < MT; ++nd) {
            float a0=0.f,a1=0.f,a2=0.f,a3=0.f,a4=0.f,a5=0.f,a6=0.f,a7=0.f;
            #pragma unroll
            for (int nb = 0; nb < KNB; ++nb) {
                const int   j  = sm_idx[nd*KNB + nb];
                const float f  = feats[(size_t)j * CC + c0 + t];
                const v4f   e0 = ew4[(nd*KNB + nb)*2 + 0];
                const v4f   e1 = ew4[(nd*KNB + nb)*2 + 1];
                a0 += e0.x * f; a1 += e0.y * f; a2 += e0.z * f; a3 += e0.w * f;
                a4 += e1.x * f; a5 += e1.y * f; a6 += e1.z * f; a7 += e1.w * f;
            }
            sm_agg[(0*MT + nd)*SP + t] = a0;
            sm_agg[(1*MT + nd)*SP + t] = a1;
            sm_agg[(2*MT + nd)*SP + t] = a2;
            sm_agg[(3*MT + nd)*SP + t] = a3;
            sm_agg[(4*MT + nd)*SP + t] = a4;
            sm_agg[(5*MT + nd)*SP + t] = a5;
            sm_agg[(6*MT + nd)*SP + t] = a6;
            sm_agg[(7*MT + nd)*SP + t] = a7;
        }
        __syncthreads();

        // ---- fp32 WMMA: [16 x CHUNK] (LDS) x [CHUNK x 128] (global) ----
        const float* aL = sm_agg + (wave*MT + colL)*SP + koff;
        #pragma unroll 4
        for (int kk = 0; kk < CHUNK/4; ++kk) {
            const v2f a = *(const v2f*)(aL + 4*kk);            // A: M=colL, K=koff..koff+1
            #pragma unroll
            for (int dt = 0; dt < 8; ++dt) {
                const v2f b = *(const v2f*)(bbase + (size_t)dt*16*CC + c0 + 4*kk);
                acc[dt] = __builtin_amdgcn_wmma_f32_16x16x4_f32(
                    /*neg_a=*/false, a, /*neg_b=*/false, b,
                    /*c_mod=*/(short)0, acc[dt],
                    /*reuse_a=*/false, /*reuse_b=*/false);
            }
        }
        __syncthreads();   // protect sm_agg before next chunk overwrites it
    }

    // ---------------- store D tiles with ReLU ----------------
    // D layout: VGPR v -> row M = v + 8*half, col = colL
    const int kbase = wave * DK;
    #pragma unroll
    for (int dt = 0; dt < 8; ++dt) {
        #pragma unroll
        for (int v = 0; v < 8; ++v) {
            const int m = v + half*8;
            const float val = acc[dt][v];
            out[(size_t)(node0 + m)*(KER*DK) + kbase + dt*16 + colL] = fmaxf(val, 0.f);
        }
    }
}

extern "C" void kernel_launch(void* const* d_in, const int* in_sizes, int n_in,
                              void* d_out, int out_size, void* d_ws, size_t ws_size,
                              hipStream_t stream) {
    (void)in_sizes; (void)n_in; (void)d_ws; (void)ws_size; (void)out_size;
    const float* feats  = (const float*)d_in[0];
    const float* centre = (const float*)d_in[1];
    const int*   nidx   = (const int*)  d_in[2];
    const float* gwt    = (const float*)d_in[3];
    const float* mr     = (const float*)d_in[4];
    const float* mt     = (const float*)d_in[5];
    const float* pr     = (const float*)d_in[6];
    const float* pt     = (const float*)d_in[7];
    const float* cw     = (const float*)d_in[8];
    float*       out    = (float*)d_out;

    const size_t shmem = (size_t)SMEM_FLOATS * sizeof(float);   // 142336 B < 320 KB/WGP
    hipFuncSetAttribute(reinterpret_cast<const void*>(cgs_graphconv_wmma),
                        hipFuncAttributeMaxDynamicSharedMemorySize, (int)shmem);
    cgs_graphconv_wmma<<<NBLK, 256, shmem, stream>>>(
        feats, centre, nidx, gwt, mr, mt, pr, pt, cw, out);
}